// IntensitiyTransform_3616362463510
// MI455X (gfx1250) — compile-verified
//
#include <hip/hip_runtime.h>

// out[b,h,w,c] = it[b, clamp(rint(255*im[b,h,w,c]), 0, 255), c]
// Pure bandwidth kernel: ~101 MB of streaming traffic -> ~4.3 us at 23.3 TB/s.
// Strategy: per-block LDS-resident LUT (3 KB, transposed [c][256]), B128
// non-temporal global loads/stores, rint->v_rndne for half-to-even rounding.

typedef float v4f __attribute__((ext_vector_type(4)));

constexpr int kB = 16;
constexpr int kH = 512;
constexpr int kW = 512;
constexpr int kC = 3;
constexpr int kI = 256;
constexpr int kHWC = kH * kW * kC;                        // 786432 floats / image
constexpr int kThreads = 256;                             // 8 wave32s
constexpr int kF4PerThread = 8;
constexpr int kF4PerBlock = kThreads * kF4PerThread;      // 2048 float4 = 32 KB
constexpr int kBlocksPerImage = kHWC / (4 * kF4PerBlock); // 96 (exact)

static_assert(kBlocksPerImage * kF4PerBlock * 4 == kHWC, "exact tiling");

__device__ __forceinline__ int quant_idx(float x) {
  // jnp.round == round-half-to-even == rintf in default RNE mode (v_rndne_f32)
  int i = (int)__builtin_rintf(255.0f * x);
  i = i < 0 ? 0 : (i > 255 ? 255 : i);
  return i;
}

__global__ __launch_bounds__(kThreads) void
IntensityTransform_lut_kernel(const float* __restrict__ im,
                              const float* __restrict__ it,
                              float* __restrict__ out) {
  // LUT for this image, transposed to [c][256] so channel base is a constant add.
  __shared__ float lut[kC * kI];  // 3 KB of the 320 KB/WGP LDS

  const int b = blockIdx.y;
  const float* itb = it + (size_t)b * (kI * kC);  // source layout [256][3]
  for (int i = threadIdx.x; i < kI * kC; i += kThreads) {
    const int idx = i / kC;
    const int c   = i - idx * kC;
    lut[c * kI + idx] = itb[i];
  }
  __syncthreads();

  const float* imb  = im  + (size_t)b * kHWC;
  float*       outb = out + (size_t)b * kHWC;
  const int base4 = blockIdx.x * kF4PerBlock;

  // Touch the far end of this block's input range early (global_prefetch_b8).
  __builtin_prefetch(imb + (size_t)(base4 + kF4PerBlock - kThreads) * 4, 0, 1);

#pragma unroll
  for (int k = 0; k < kF4PerThread; ++k) {
    const int f4 = base4 + k * kThreads + threadIdx.x;  // coalesced float4 index

    // Single-use stream: non-temporal B128 load (TH=NT), don't pollute L2.
    v4f v = __builtin_nontemporal_load((const v4f*)imb + f4);

    // float offset is 4*f4; (4*f4) % 3 == f4 % 3 since 4 == 1 (mod 3).
    const int c0 = f4 % 3;
    const int c1 = (c0 == 2) ? 0 : c0 + 1;
    const int c2 = (c1 == 2) ? 0 : c1 + 1;

    v4f o;
    o.x = lut[c0 * kI + quant_idx(v.x)];
    o.y = lut[c1 * kI + quant_idx(v.y)];
    o.z = lut[c2 * kI + quant_idx(v.z)];
    o.w = lut[c0 * kI + quant_idx(v.w)];

    // Single-use stream: non-temporal B128 store.
    __builtin_nontemporal_store(o, (v4f*)outb + f4);
  }
}

extern "C" void kernel_launch(void* const* d_in, const int* in_sizes, int n_in,
                              void* d_out, int out_size, void* d_ws, size_t ws_size,
                              hipStream_t stream) {
  const float* im = (const float*)d_in[0];  // [16,512,512,3] f32
  const float* it = (const float*)d_in[1];  // [16,256,3]     f32
  float* out = (float*)d_out;               // [16,512,512,3] f32
  (void)in_sizes; (void)n_in; (void)out_size; (void)d_ws; (void)ws_size;

  dim3 grid(kBlocksPerImage, kB, 1);
  IntensityTransform_lut_kernel<<<grid, kThreads, 0, stream>>>(im, it, out);
}